// WEFPositionalEncoding_45956150067261
// MI455X (gfx1250) — compile-verified
//
#include <hip/hip_runtime.h>

// Problem constants (fixed by the reference file)
static constexpr int B_  = 64;
static constexpr int H_  = 32;
static constexpr int W_  = 32;
static constexpr int S_  = 1 + H_ * W_;   // 1025
static constexpr int D_  = 768;
static constexpr int HW_ = H_ * W_;       // 1024
#define OMEGA1F 2.62205755429212f

typedef __attribute__((ext_vector_type(2))) float v2f;
typedef __attribute__((ext_vector_type(8))) float v8f;

__device__ __forceinline__ float clamp_pm(float v, float lim) {
    return fminf(fmaxf(v, -lim), lim);
}

// ---------------------------------------------------------------------------
// Kernel 1a: Weierstrass wp / wp' -> tanh features.  One wave32 per grid
// point; each lane sums ~20 lattice terms, then a full cross-lane xor-shuffle
// reduction.  1024 waves -> fills the GPU, sub-microsecond.
// ---------------------------------------------------------------------------
__global__ void wp_feats_kernel(const float* __restrict__ log_alpha_scale,
                                const float* __restrict__ alpha_learn,
                                float* __restrict__ feats /* [1024][4] */) {
    int gid   = blockIdx.x * blockDim.x + threadIdx.x;
    int point = gid >> 5;
    int lane  = gid & 31;
    if (point >= HW_) return;

    // omega3' = clip(softplus(alpha_learn), 0.02, 8.0)
    float om3 = fminf(fmaxf(log1pf(expf(alpha_learn[0])), 0.02f), 8.0f);

    int   row = point >> 5;          // /32
    int   col = point & 31;          // %32
    float u  = (col + 0.5f) * (1.0f / W_);
    float v  = (row + 0.5f) * (1.0f / H_);
    float zr = u * (2.0f * OMEGA1F) * 0.4f;
    float zi = v * (2.0f * om3) * 0.4f;

    float s_wpr = 0.f, s_wpi = 0.f, s_wppr = 0.f, s_wppi = 0.f;
    for (int idx = lane; idx < 625; idx += 32) {
        if (idx == 312) continue;                       // (m,n) == (0,0)
        int   m  = idx / 25 - 12;
        int   n  = idx % 25 - 12;
        float wr = 2.0f * m * OMEGA1F;
        float wi = 2.0f * n * OMEGA1F;
        // 1/w^2
        float rw  = wr * wr + wi * wi;
        float irw = 1.0f / rw;
        float ar  = wr * irw, ai = -wi * irw;           // 1/w
        float w2r = ar * ar - ai * ai;
        float w2i = 2.0f * ar * ai;                     // 1/w^2
        // 1/d^2, 1/d^3
        float dr = zr - wr, di = zi - wi;
        float rd = dr * dr + di * di;
        if (rd > (1.5e-7f * 1.5e-7f)) {
            float ird = 1.0f / rd;
            float br  = dr * ird, bi = -di * ird;       // 1/d
            float d2r = br * br - bi * bi;
            float d2i = 2.0f * br * bi;                 // 1/d^2
            float d3r = d2r * br - d2i * bi;
            float d3i = d2r * bi + d2i * br;            // 1/d^3
            s_wpr  += clamp_pm(d2r - w2r, 5000.0f);
            s_wpi  += clamp_pm(d2i - w2i, 5000.0f);
            s_wppr += clamp_pm(-2.0f * d3r, 5000.0f);
            s_wppi += clamp_pm(-2.0f * d3i, 5000.0f);
        }
    }
    #pragma unroll
    for (int off = 16; off >= 1; off >>= 1) {
        s_wpr  += __shfl_xor(s_wpr,  off, 32);
        s_wpi  += __shfl_xor(s_wpi,  off, 32);
        s_wppr += __shfl_xor(s_wppr, off, 32);
        s_wppi += __shfl_xor(s_wppi, off, 32);
    }
    if (lane == 0) {
        float rz  = zr * zr + zi * zi;
        float irz = 1.0f / rz;
        float cr  = zr * irz, ci = -zi * irz;           // 1/z
        float z2r = cr * cr - ci * ci, z2i = 2.0f * cr * ci;
        float z3r = z2r * cr - z2i * ci, z3i = z2r * ci + z2i * cr;
        float wpr  = clamp_pm(z2r + s_wpr, 10000.0f);
        float wpi  = clamp_pm(z2i + s_wpi, 10000.0f);
        float wppr = clamp_pm(-2.0f * z3r + s_wppr, 10000.0f);
        float wppi = clamp_pm(-2.0f * z3i + s_wppi, 10000.0f);
        float a = fminf(fmaxf(expf(log_alpha_scale[0]), 0.002f), 0.8f);
        feats[point * 4 + 0] = tanhf(a * wpr);
        feats[point * 4 + 1] = tanhf(a * wpi);
        feats[point * 4 + 2] = tanhf(a * wppr);
        feats[point * 4 + 3] = tanhf(a * wppi);
    }
}

// ---------------------------------------------------------------------------
// Kernel 1b: projection via V_WMMA_F32_16X16X4_F32 (exact shape match:
// [16x4] x [4x16] f32) + LayerNorm + gamma/beta + pos_scale fold.
// One block = one 16-row M-tile; 8 waves x 6 N-tiles cover all 768 columns.
// ---------------------------------------------------------------------------
#define LDS_STRIDE 772   // 768 + 4 pad: breaks bank conflicts on column sweeps
__global__ void proj_ln_kernel(const float* __restrict__ feats,
                               const float* __restrict__ Wp,     // [4][768]
                               const float* __restrict__ bias,   // [768]
                               const float* __restrict__ gamma,  // [768]
                               const float* __restrict__ beta,   // [768]
                               const float* __restrict__ pos_scale,
                               float* __restrict__ pe_full /* [1025][768] */) {
    __shared__ float lds[16 * LDS_STRIDE];
    const int tid  = threadIdx.x;
    const int wv   = tid >> 5;
    const int lane = tid & 31;
    const int m0   = blockIdx.x * 16;

    // A fragment: 16x4 f32, 2 VGPRs/lane.
    // lanes 0-15: {K=0,K=1} of row M=lane; lanes 16-31: {K=2,K=3} of M=lane-16
    v2f a;
    {
        const float* fr = feats + (m0 + (lane & 15)) * 4 + ((lane < 16) ? 0 : 2);
        a.x = fr[0];
        a.y = fr[1];
    }
    v8f cz = {};

    for (int j = 0; j < 6; ++j) {
        const int nt  = wv * 6 + j;
        const int col = nt * 16 + (lane & 15);
        // B fragment: 4x16 f32 (mirrored layout).
        v2f b;
        if (lane < 16) { b.x = Wp[0 * D_ + col]; b.y = Wp[1 * D_ + col]; }
        else           { b.x = Wp[2 * D_ + col]; b.y = Wp[3 * D_ + col]; }

        v8f d = __builtin_amdgcn_wmma_f32_16x16x4_f32(
            /*neg_a=*/false, a, /*neg_b=*/false, b,
            /*c_mod=*/(short)0, cz, /*reuse_a=*/false, /*reuse_b=*/false);

        const float bs    = bias[col];
        const int   rbase = (lane < 16) ? 0 : 8;   // D layout: lanes16-31 -> M=r+8
        #pragma unroll
        for (int r = 0; r < 8; ++r)
            lds[(rbase + r) * LDS_STRIDE + col] = d[r] + bs;
    }
    __syncthreads();

    // LayerNorm: 16 threads per row (stay within one 16-lane shuffle group).
    const int row = tid >> 4;
    const int sub = tid & 15;
    float s = 0.f, q = 0.f;
    for (int c = sub; c < D_; c += 16) {
        float v = lds[row * LDS_STRIDE + c];
        s += v;
        q += v * v;
    }
    #pragma unroll
    for (int off = 8; off >= 1; off >>= 1) {
        s += __shfl_xor(s, off, 32);
        q += __shfl_xor(q, off, 32);
    }
    const float mu   = s * (1.0f / D_);
    const float var  = q * (1.0f / D_) - mu * mu;
    const float rstd = rsqrtf(var + 1e-5f);
    const float ps   = pos_scale[0];
    float* outrow = pe_full + (size_t)(1 + m0 + row) * D_;
    for (int c = sub; c < D_; c += 16) {
        float v = lds[row * LDS_STRIDE + c];
        outrow[c] = ((v - mu) * rstd * gamma[c] + beta[c]) * ps;
    }
}

// ---------------------------------------------------------------------------
// Kernel 1c: CLS row of pe table (row 0), pos_scale folded in.
// ---------------------------------------------------------------------------
__global__ void cls_kernel(const float* __restrict__ cls_pos,
                           const float* __restrict__ pos_scale,
                           float* __restrict__ pe_full) {
    int i = blockIdx.x * blockDim.x + threadIdx.x;
    if (i < D_) pe_full[i] = cls_pos[i] * pos_scale[0];
}

// ---------------------------------------------------------------------------
// Kernel 2: out = x + pe (pe already scaled).  The roofline kernel:
// 403 MB HBM traffic -> ~17 us at 23.3 TB/s.  float4 (B128) coalesced
// streams; pe table (3.15 MB) stays resident in the 192 MB L2.
// ---------------------------------------------------------------------------
__global__ void add_pe_kernel(const float4* __restrict__ x,
                              const float4* __restrict__ pe,
                              float4* __restrict__ out,
                              int total4, int per_batch4) {
    int i = blockIdx.x * blockDim.x + threadIdx.x;
    if (i >= total4) return;
    int p = i % per_batch4;                 // magic-mul, no real div
    float4 xv = x[i];
    float4 pv = pe[p];
    float4 o;
    o.x = xv.x + pv.x;
    o.y = xv.y + pv.y;
    o.z = xv.z + pv.z;
    o.w = xv.w + pv.w;
    out[i] = o;
}

extern "C" void kernel_launch(void* const* d_in, const int* in_sizes, int n_in,
                              void* d_out, int out_size, void* d_ws, size_t ws_size,
                              hipStream_t stream) {
    const float* x           = (const float*)d_in[0];
    // d_in[1] = h, d_in[2] = w : fixed at 32 by the reference; unused.
    const float* log_alpha   = (const float*)d_in[3];
    const float* alpha_learn = (const float*)d_in[4];
    const float* Wp          = (const float*)d_in[5];
    const float* bp          = (const float*)d_in[6];
    const float* gamma       = (const float*)d_in[7];
    const float* beta        = (const float*)d_in[8];
    const float* cls         = (const float*)d_in[9];
    const float* ps          = (const float*)d_in[10];
    float* out = (float*)d_out;

    // Workspace: pe table [1025][768] f32, then feats [1024][4] f32.
    float* pe_full = (float*)d_ws;
    float* feats   = pe_full + (size_t)S_ * D_;

    // 1a: features (1024 points x 1 wave each = 32768 threads)
    wp_feats_kernel<<<(HW_ * 32 + 255) / 256, 256, 0, stream>>>(log_alpha, alpha_learn, feats);
    // 1c: CLS row
    cls_kernel<<<(D_ + 255) / 256, 256, 0, stream>>>(cls, ps, pe_full);
    // 1b: WMMA projection + LayerNorm (64 M-tiles)
    proj_ln_kernel<<<HW_ / 16, 256, 0, stream>>>(feats, Wp, bp, gamma, beta, ps, pe_full);
    // 2: bandwidth-bound broadcast add
    const int per_batch4 = S_ * (D_ / 4);          // 196800
    const int total4     = B_ * per_batch4;        // 12,595,200
    add_pe_kernel<<<(total4 + 255) / 256, 256, 0, stream>>>(
        (const float4*)x, (const float4*)pe_full, (float4*)out, total4, per_batch4);
}